// IDSCT2_1133871366823
// MI455X (gfx1250) — compile-verified
//
#include <hip/hip_runtime.h>

// ---------------------------------------------------------------------------
// idsct2: out[b,u,v] = sum_{p,q} x[b,p,q] * sin(pi(2u+1)p/2M) * cos(pi(2v+1)q/2N)
// Stage 0: basis f16 matrices + x -> f16.
// Stage 1: Tt[b][v][p] = sum_q Xh[b][p][q] * Cmat[v][q]   (WMMA, transposed store)
// Stage 2: out[b][u][v] = sum_p Smat[u][p] * Tt[b][v][p]  (WMMA, f32 store)
// Global->LDS staging uses CDNA5 async-to-LDS (ASYNCcnt); WMMAs are issued as
// tied-operand inline asm so accumulation is in-place (no phi-copy v_movs).
// ---------------------------------------------------------------------------

typedef __attribute__((ext_vector_type(16))) _Float16 v16h;
typedef __attribute__((ext_vector_type(8)))  float    v8f;

union F16Frag { v16h v; uint4 q[2]; };

static __device__ __forceinline__ unsigned pack2(float a, float b) {
    union { _Float16 h[2]; unsigned u; } t;
    t.h[0] = (_Float16)a;
    t.h[1] = (_Float16)b;
    return t.u;
}

// GVS-form async copy: copies 16B/lane from global sbase plus offset voff to the LDS address in ldsoff, ASYNCcnt.
#define ASYNC_LDS_B128(ldsoff, voff, sbase)                                    \
    asm volatile("global_load_async_to_lds_b128 %0, %1, %2"                    \
                 :: "v"(ldsoff), "v"(voff), "s"(sbase) : "memory")

#define WAIT_ASYNC0() asm volatile("s_wait_asynccnt 0x0" ::: "memory")

// In-place D = A*B + D (D tied to C -> same even-aligned VGPR tuple).
#define WMMA_F16(accv, av, bv)                                                 \
    asm("v_wmma_f32_16x16x32_f16 %0, %1, %2, %0"                               \
        : "+v"(accv) : "v"(av), "v"(bv))

// ---------------------------------------------------------------------------
// Basis: Cmat[r*2048+c] = cos(pi*(2r+1)*c/4096), Smat = sin(...).
// Integer phase reduction mod 8192 keeps the sincos argument in [0, 2pi).
// ---------------------------------------------------------------------------
__global__ void idsct2_fill_basis(_Float16* __restrict__ Cmat,
                                  _Float16* __restrict__ Smat) {
    int idx = blockIdx.x * 256 + threadIdx.x;   // 0 .. 2048*2048-1
    int c = idx & 2047;
    int r = idx >> 11;
    int t = ((2 * r + 1) * c) & 8191;
    float a = (float)t * 7.6699039394282060e-4f;  // pi/4096
    float s, co;
    __sincosf(a, &s, &co);
    Cmat[idx] = (_Float16)co;
    Smat[idx] = (_Float16)s;
}

// x (f32) -> Xh (f16), one batch (2048x2048), 8 elements/thread.
__global__ void idsct2_convert_x(const float* __restrict__ x,
                                 _Float16* __restrict__ Xh) {
    size_t i = ((size_t)blockIdx.x * 256 + threadIdx.x) * 8;
    float4 a = *(const float4*)(x + i);
    float4 b = *(const float4*)(x + i + 4);
    uint4 u;
    u.x = pack2(a.x, a.y);
    u.y = pack2(a.z, a.w);
    u.z = pack2(b.x, b.y);
    u.w = pack2(b.z, b.w);
    *(uint4*)(Xh + i) = u;
}

// ---------------------------------------------------------------------------
// Fragment loads + 8 WMMAs for one 32-wide K-step of a 128x128 block tile.
// LDS tile layout: 128 rows x 80-byte pitch (64B data + 16B pad), A then B.
// ---------------------------------------------------------------------------
static __device__ __forceinline__ void wmma_step(const unsigned char* base,
                                                 int wm, int wn, int l15, int lhi,
                                                 v8f acc[2][4]) {
    F16Frag afrag[2], bfrag[4];
#pragma unroll
    for (int i = 0; i < 2; ++i) {
        // A 16x32 f16: lanes<16 hold K{0..7,16..23}, lanes>=16 hold K{8..15,24..31}
        int row = wm * 32 + i * 16 + l15;
        const unsigned char* p = base + row * 80 + lhi * 16;
        afrag[i].q[0] = *(const uint4*)p;
        afrag[i].q[1] = *(const uint4*)(p + 32);
    }
#pragma unroll
    for (int j = 0; j < 4; ++j) {
        // B 32x16 f16 (n-major rows): lanes<16 hold K0..15, lanes>=16 hold K16..31
        int row = wn * 64 + j * 16 + l15;
        const unsigned char* p = base + 10240 + row * 80 + lhi * 32;
        bfrag[j].q[0] = *(const uint4*)p;
        bfrag[j].q[1] = *(const uint4*)(p + 16);
    }
#pragma unroll
    for (int i = 0; i < 2; ++i)
#pragma unroll
        for (int j = 0; j < 4; ++j)
            WMMA_F16(acc[i][j], afrag[i].v, bfrag[j].v);
}

// ---------------------------------------------------------------------------
// D[m,n] = sum_k A[m,k] * B[n,k]   (A row-major f16, B n-major f16, K=2048)
// Block 128x128, K-step 32, 8 waves (4 along M x 2 along N), wave = 32x64.
// STAGE1: D stored transposed f16 (Dt16[n*2048+m]); STAGE2: f32 direct.
// ---------------------------------------------------------------------------
template <bool STAGE1>
__global__ __launch_bounds__(256)
void idsct2_gemm(const _Float16* __restrict__ A,     // row-major, pre-offset
                 const _Float16* __restrict__ Bn,    // n-major (stage2: batched)
                 _Float16*       __restrict__ Dt16,  // stage1 out (pre-offset)
                 float*          __restrict__ Df32)  // stage2 out (batched)
{
    const int tid  = threadIdx.x;
    const int lane = tid & 31;
    const int wave = tid >> 5;
    const int wm   = wave & 3;
    const int wn   = wave >> 2;
    const int l15  = lane & 15;
    const int lhi  = lane >> 4;

    const int bm = blockIdx.y * 128;
    const int bn = blockIdx.x * 128;
    const int bz = blockIdx.z;

    __shared__ __align__(16) unsigned char smem[2 * (10240 + 10240)];

    const _Float16* Bg = STAGE1 ? Bn : (Bn + (size_t)bz * 2048 * 2048);

    // Async staging: 256 threads x 4 chunks of 16B cover A(8KB)+B(8KB) per tile.
    const int crow = tid >> 2;          // 0..63
    const int cq   = tid & 3;           // 16B chunk within a 64B row
    const unsigned voff0 = (unsigned)(crow * 4096 + cq * 16);      // rows 0..63
    const unsigned voff1 = voff0 + 64u * 4096u;                    // rows 64..127
    const unsigned ldsbase =
        (unsigned)(size_t)smem + (unsigned)(crow * 80 + cq * 16);  // buffer 0

    unsigned long long sA = (unsigned long long)(size_t)A  + (size_t)bm * 4096;
    unsigned long long sB = (unsigned long long)(size_t)Bg + (size_t)bn * 4096;

    v8f acc[2][4];
#pragma unroll
    for (int i = 0; i < 2; ++i)
#pragma unroll
        for (int j = 0; j < 4; ++j) {
            v8f z = {};
            acc[i][j] = z;
        }

    // Prologue: tile 0 -> buffer 0.
    ASYNC_LDS_B128(ldsbase,          voff0, sA);
    ASYNC_LDS_B128(ldsbase +  5120u, voff1, sA);
    ASYNC_LDS_B128(ldsbase + 10240u, voff0, sB);
    ASYNC_LDS_B128(ldsbase + 15360u, voff1, sB);

    for (int kt = 0; kt < 64; ++kt) {
        WAIT_ASYNC0();        // tile kt resident in buffer kt&1 (this wave's part)
        __syncthreads();      // all waves' parts resident; prior reads retired
        if (kt < 63) {
            sA += 64;         // advance K by 32 f16
            sB += 64;
            unsigned lb = ldsbase + (((kt + 1) & 1) ? 20480u : 0u);
            ASYNC_LDS_B128(lb,          voff0, sA);
            ASYNC_LDS_B128(lb +  5120u, voff1, sA);
            ASYNC_LDS_B128(lb + 10240u, voff0, sB);
            ASYNC_LDS_B128(lb + 15360u, voff1, sB);
        }
        wmma_step(smem + (kt & 1) * 20480, wm, wn, l15, lhi, acc);
    }

    // Epilogue. D frag: lanes 0-15 -> N=lane, M=r; lanes 16-31 -> N=lane-16, M=8+r.
    if (STAGE1) {
#pragma unroll
        for (int i = 0; i < 2; ++i)
#pragma unroll
            for (int j = 0; j < 4; ++j) {
                int m0 = bm + wm * 32 + i * 16 + lhi * 8;
                int n  = bn + wn * 64 + j * 16 + l15;
                uint4 u;
                u.x = pack2(acc[i][j][0], acc[i][j][1]);
                u.y = pack2(acc[i][j][2], acc[i][j][3]);
                u.z = pack2(acc[i][j][4], acc[i][j][5]);
                u.w = pack2(acc[i][j][6], acc[i][j][7]);
                *(uint4*)(Dt16 + (size_t)n * 2048 + m0) = u;   // transposed store
            }
    } else {
        float* O = Df32 + (size_t)bz * 2048 * 2048;
#pragma unroll
        for (int i = 0; i < 2; ++i)
#pragma unroll
            for (int j = 0; j < 4; ++j) {
                int m0 = bm + wm * 32 + i * 16 + lhi * 8;
                int n  = bn + wn * 64 + j * 16 + l15;
#pragma unroll
                for (int r = 0; r < 8; ++r)
                    O[(size_t)(m0 + r) * 2048 + n] = acc[i][j][r];
            }
    }
}

// ---------------------------------------------------------------------------
// Workspace (56 MB):
//   [ 0, 8 MB)  Cmat f16 [v][q]
//   [ 8,16 MB)  Smat f16 [u][p]
//   [16,24 MB)  Xh   f16 [p][q]      (one batch, reused)
//   [24,56 MB)  Tt   f16 [b][v][p]   (stage-1 output, transposed)
// ---------------------------------------------------------------------------
extern "C" void kernel_launch(void* const* d_in, const int* in_sizes, int n_in,
                              void* d_out, int out_size, void* d_ws, size_t ws_size,
                              hipStream_t stream) {
    (void)in_sizes; (void)n_in; (void)out_size; (void)ws_size;

    const float* x   = (const float*)d_in[0];
    float*       out = (float*)d_out;
    char*        ws  = (char*)d_ws;

    _Float16* Cmat = (_Float16*)(ws);
    _Float16* Smat = (_Float16*)(ws + (size_t) 8 * 1024 * 1024);
    _Float16* Xh   = (_Float16*)(ws + (size_t)16 * 1024 * 1024);
    _Float16* Tt   = (_Float16*)(ws + (size_t)24 * 1024 * 1024);

    idsct2_fill_basis<<<(2048 * 2048) / 256, 256, 0, stream>>>(Cmat, Smat);

    dim3 block(256);
    for (int b = 0; b < 4; ++b) {
        const size_t bo = (size_t)b * 2048 * 2048;
        idsct2_convert_x<<<(2048 * 2048) / (256 * 8), block, 0, stream>>>(x + bo, Xh);
        // Tt[b][v][p] = sum_q Xh[p][q] * Cmat[v][q]
        idsct2_gemm<true><<<dim3(16, 16, 1), block, 0, stream>>>(
            Xh, Cmat, Tt + bo, nullptr);
    }
    // out[b][u][v] = sum_p Smat[u][p] * Tt[b][v][p]
    idsct2_gemm<false><<<dim3(16, 16, 4), block, 0, stream>>>(
        Smat, Tt, nullptr, out);
}